// CanineEmbeddings_49675591745547
// MI455X (gfx1250) — compile-verified
//
#include <hip/hip_runtime.h>
#include <stdint.h>
#include <stddef.h>

// ---- problem constants (from reference) ----
#define NUM_HASHES   8
#define NUM_BUCKETS  16384
#define HIDDEN       768
#define SHARD        96            // HIDDEN / NUM_HASHES
#define BATCH        8
#define SEQ          8192
#define TOKENS       (BATCH * SEQ) // 65536
#define LN_EPS       1e-6f

// 16B vector type (ext_vector so clang supports vector<->scalar arithmetic)
typedef float v4f __attribute__((ext_vector_type(4)));

// One token per wave32; 8 waves (256 threads) per block.
// Per token: 3 KB = 192 x 16B chunks; each lane owns 6 chunks (chunk = c*32+lane).
// Shard rows are 384 B = 24 chunks, so a 16B chunk never straddles a shard row.
//
// For unrolled chunk group c, chunk = c*32 + lane spans exactly two shard
// indices, split at a fixed lane threshold -> h and the hash prime are
// compile-time constant selects (no table in memory):
//   c=0: chunks   0..31 : h = lane<24 ? 0 : 1   prime 31 / 43
//   c=1: chunks  32..63 : h = lane<16 ? 1 : 2   prime 43 / 59
//   c=2: chunks  64..95 : h = lane< 8 ? 2 : 3   prime 59 / 61
//   c=3: chunks  96..127: h = lane<24 ? 4 : 5   prime 73 / 97
//   c=4: chunks 128..159: h = lane<16 ? 5 : 6   prime 97 / 103
//   c=5: chunks 160..191: h = lane< 8 ? 6 : 7   prime 103/ 113
__global__ __launch_bounds__(256)
void canine_hash_emb_ln_kernel(const int* __restrict__ ids,
                               const float* __restrict__ tables,
                               const float* __restrict__ gamma,
                               const float* __restrict__ beta,
                               float* __restrict__ out)
{
    __shared__ v4f s_tok[8 * (HIDDEN / 4)];   // 8 waves x 192 float4 = 24 KB gather staging
    __shared__ v4f s_gamma[HIDDEN / 4];       // 3 KB
    __shared__ v4f s_beta[HIDDEN / 4];        // 3 KB

    const int tid  = threadIdx.x;
    const int lane = tid & 31;
    const int wave = tid >> 5;

    // Cache LayerNorm affine params in LDS (768 floats each; 3 per thread).
    {
        float* sg = (float*)s_gamma;
        float* sb = (float*)s_beta;
#pragma unroll
        for (int i = 0; i < 3; ++i) {
            const int e = tid + i * 256;
            sg[e] = gamma[e];
            sb[e] = beta[e];
        }
    }

    const int token = blockIdx.x * 8 + wave;
    const unsigned idp1 = (unsigned)(ids[token] + 1);

    // LDS byte offset of this wave's staging buffer (low 32 bits of the
    // generic shared-space address are the LDS offset on amdgcn).
    v4f* tok = &s_tok[wave * (HIDDEN / 4)];
    const uint32_t lds_base = (uint32_t)(uintptr_t)tok;

    // ---- issue 6 async gather loads per lane: global -> LDS (ASYNCcnt path) ----
    {
        constexpr int      kThr[6] = {24, 16, 8, 24, 16, 8};
        constexpr unsigned kPlo[6] = {31u, 43u, 59u, 73u, 97u, 103u};
        constexpr unsigned kPhi[6] = {43u, 59u, 61u, 97u, 103u, 113u};
        constexpr unsigned kHlo[6] = {0u, 1u, 2u, 4u, 5u, 6u};
#pragma unroll
        for (int c = 0; c < 6; ++c) {
            const int      chunk  = c * 32 + lane;               // 0..191
            const bool     low    = lane < kThr[c];
            const unsigned h      = kHlo[c] + (low ? 0u : 1u);   // shard / hash idx
            const unsigned prime  = low ? kPlo[c] : kPhi[c];
            const int      within = (chunk - (int)h * 24) * 16;  // byte off in row
            const unsigned bucket = (idp1 * prime) & (NUM_BUCKETS - 1);
            const char* src = (const char*)tables +
                              ((size_t)h * NUM_BUCKETS + bucket) * (SHARD * 4) + within;
            const uint32_t dst = lds_base + (uint32_t)chunk * 16u;
            asm volatile("global_load_async_to_lds_b128 %0, %1, off"
                         :
                         : "v"(dst), "v"(src)
                         : "memory");
        }
    }

    __syncthreads();  // publish s_gamma/s_beta (staging buffers are wave-private)

    // Wait for this wave's async gathers to land in LDS.
#if __has_builtin(__builtin_amdgcn_s_wait_asynccnt)
    __builtin_amdgcn_s_wait_asynccnt(0);
    asm volatile("" ::: "memory");
#else
    asm volatile("s_wait_asynccnt 0" ::: "memory");
#endif

    // ---- read back 24 floats per lane and reduce for LayerNorm ----
    v4f x[6];
    float sum = 0.f, sumsq = 0.f;
#pragma unroll
    for (int c = 0; c < 6; ++c) {
        const int chunk = c * 32 + lane;
        x[c] = tok[chunk];
        sum   += x[c].x + x[c].y + x[c].z + x[c].w;
        sumsq += x[c].x * x[c].x + x[c].y * x[c].y +
                 x[c].z * x[c].z + x[c].w * x[c].w;
    }

    // wave32 butterfly reduction
#pragma unroll
    for (int off = 16; off >= 1; off >>= 1) {
        sum   += __shfl_xor(sum, off, 32);
        sumsq += __shfl_xor(sumsq, off, 32);
    }

    const float mean = sum * (1.0f / (float)HIDDEN);
    const float var  = sumsq * (1.0f / (float)HIDDEN) - mean * mean;
    const float rstd = rsqrtf(var + LN_EPS);

    // ---- normalize + affine, stream out with non-temporal 16B stores ----
    float* outTok = out + (size_t)token * HIDDEN;
#pragma unroll
    for (int c = 0; c < 6; ++c) {
        const int chunk = c * 32 + lane;
        const v4f g = s_gamma[chunk];
        const v4f b = s_beta[chunk];
        v4f y = (x[c] - mean) * rstd;
        y = y * g + b;
        __builtin_nontemporal_store(y, (v4f*)(outTok + chunk * 4));
    }
}

extern "C" void kernel_launch(void* const* d_in, const int* in_sizes, int n_in,
                              void* d_out, int out_size, void* d_ws, size_t ws_size,
                              hipStream_t stream) {
    (void)in_sizes; (void)n_in; (void)d_ws; (void)ws_size; (void)out_size;
    const int*   ids    = (const int*)d_in[0];
    const float* tables = (const float*)d_in[1];
    const float* gamma  = (const float*)d_in[2];
    const float* beta   = (const float*)d_in[3];
    float*       out    = (float*)d_out;

    dim3 grid(TOKENS / 8);   // 8192 blocks
    dim3 block(256);         // 8 wave32 per block, 1 token per wave
    hipLaunchKernelGGL(canine_hash_emb_ln_kernel, grid, block, 0, stream,
                       ids, tables, gamma, beta, out);
}